// GOKUnet_23244363006422
// MI455X (gfx1250) — compile-verified
//
#include <hip/hip_runtime.h>
#include <math.h>

typedef float v2f __attribute__((ext_vector_type(2)));
typedef float v8f __attribute__((ext_vector_type(8)));
typedef int gvec4 __attribute__((vector_size(16)));   // matches builtin's <4 x i32>

#define BB   64
#define TT   50
#define IMG  16384
#define ZD   16
#define THD  4
#define AD   2
#define EH   256
#define OHID 128
#define DH   256
#define NSTEPS 100
#define KC   32

// ---------------------------------------------------------------------------
// CDNA5 async global->LDS copy (ASYNCcnt-tracked), with synchronous fallback.
// ---------------------------------------------------------------------------
#if defined(__gfx1250__) && __has_builtin(__builtin_amdgcn_global_load_async_to_lds_b128)
  #define USE_ASYNC 1
#else
  #define USE_ASYNC 0
#endif

#if USE_ASYNC
  #if __has_builtin(__builtin_amdgcn_s_wait_asynccnt)
    #define WAIT_ASYNC(n) __builtin_amdgcn_s_wait_asynccnt((unsigned short)(n))
  #else
    #define WAIT_ASYNC(n) asm volatile("s_wait_asynccnt %0" :: "n"(n) : "memory")
  #endif
#else
  #define WAIT_ASYNC(n) ((void)0)
#endif

static __device__ __forceinline__ void cp_async16(const float* g, float* l) {
#if USE_ASYNC
    __builtin_amdgcn_global_load_async_to_lds_b128(
        (__attribute__((address_space(1))) gvec4*)(g),
        (__attribute__((address_space(3))) gvec4*)(l),
        0, 0);
#else
    *(float4*)l = *(const float4*)g;
#endif
}

static __device__ __forceinline__ v8f wmma4(v2f a, v2f b, v8f c) {
    return __builtin_amdgcn_wmma_f32_16x16x4_f32(false, a, false, b, (short)0, c, false, false);
}

// ---------------------------------------------------------------------------
// Kernel 1: encoder GEMM + tanh + mean-pool over T.
// grid = (4 col-tiles of 64, 64 batch), block = 256 (8 waves).
// Double-buffered async global->LDS staging; fp32 WMMA 16x16x4.
// ---------------------------------------------------------------------------
__global__ __launch_bounds__(256)
void k_encoder(const float* __restrict__ x, const float* __restrict__ w,
               const float* __restrict__ bias, float* __restrict__ pooled) {
    const int nbase = blockIdx.x * 64;
    const int b     = blockIdx.y;
    const int tid   = threadIdx.x;
    const int lane  = tid & 31;
    const int wv    = tid >> 5;
    const int mtile = wv & 3;        // 4 m-tiles of 16 rows (t dimension, padded 50->64)
    const int npair = wv >> 2;       // 2 column halves of 32

    __shared__ float lA[2][64][36];  // 64 t-rows x 32 K (padded), double buffered
    __shared__ float lB[2][32][68];  // 32 K x 64 N (padded), double buffered
    __shared__ float poolbuf[64];

    if (tid < 64) poolbuf[tid] = 0.0f;

    v8f acc0 = {}; v8f acc1 = {};

    const int tA = tid >> 2, qA = tid & 3;   // A staging: 4 threads/row, 2x16B each
    const int rB = tid >> 4, cB = tid & 15;  // B staging: 16 threads/row, rows rB & rB+16
    const int tsrc = (tA < TT) ? tA : 0;     // pad rows alias row 0; masked in epilogue
    const float* gA = x + ((size_t)(b * TT + tsrc)) * IMG + qA * 8;

    auto stage = [&](int k0, int bf) {
        cp_async16(gA + k0,     &lA[bf][tA][qA * 8]);
        cp_async16(gA + k0 + 4, &lA[bf][tA][qA * 8 + 4]);
        cp_async16(w + (size_t)(k0 + rB) * EH + nbase + cB * 4,      &lB[bf][rB][cB * 4]);
        cp_async16(w + (size_t)(k0 + rB + 16) * EH + nbase + cB * 4, &lB[bf][rB + 16][cB * 4]);
    };

    const int kb   = (lane >> 4) * 2;            // lanes 16-31 hold K+2,K+3
    const int mrow = mtile * 16 + (lane & 15);
    const int nc0  = npair * 32 + (lane & 15);

    stage(0, 0);
    const int NCH = IMG / KC;
    for (int c = 0; c < NCH; ++c) {
        const int bf = c & 1;
        if (c + 1 < NCH) { stage((c + 1) * KC, bf ^ 1); WAIT_ASYNC(4); }
        else             { WAIT_ASYNC(0); }
        __syncthreads();
        #pragma unroll
        for (int kk = 0; kk < KC; kk += 4) {
            v2f a  = *(const v2f*)&lA[bf][mrow][kk + kb];
            v2f b0 = { lB[bf][kk + kb][nc0],      lB[bf][kk + kb + 1][nc0]      };
            v2f b1 = { lB[bf][kk + kb][nc0 + 16], lB[bf][kk + kb + 1][nc0 + 16] };
            acc0 = wmma4(a, b0, acc0);
            acc1 = wmma4(a, b1, acc1);
        }
        __syncthreads();
    }

    // Epilogue: tanh(+bias), mask t>=50, reduce over t into poolbuf.
    const int half  = lane >> 4;                 // C rows v (half=0) / v+8 (half=1)
    const int nl0   = npair * 32 + (lane & 15);
    const int nl1   = nl0 + 16;
    const float bb0 = bias[nbase + nl0];
    const float bb1 = bias[nbase + nl1];
    float s0 = 0.f, s1 = 0.f;
    const int rbase = mtile * 16 + half * 8;
    #pragma unroll
    for (int v = 0; v < 8; ++v) {
        if (rbase + v < TT) {
            s0 += tanhf(acc0[v] + bb0);
            s1 += tanhf(acc1[v] + bb1);
        }
    }
    atomicAdd(&poolbuf[nl0], s0);
    atomicAdd(&poolbuf[nl1], s1);
    __syncthreads();
    if (tid < 64) pooled[(size_t)b * EH + nbase + tid] = poolbuf[tid] * (1.0f / (float)TT);
}

// ---------------------------------------------------------------------------
// Kernel 2: latent heads + reparameterize + grounding (tiny, scalar).
// ---------------------------------------------------------------------------
__global__ __launch_bounds__(256)
void k_latent(const float* __restrict__ pooled,
              const float* __restrict__ w_mu_z0, const float* __restrict__ b_mu_z0,
              const float* __restrict__ w_lv_z0, const float* __restrict__ b_lv_z0,
              const float* __restrict__ w_mu_th, const float* __restrict__ b_mu_th,
              const float* __restrict__ w_lv_th, const float* __restrict__ b_lv_th,
              const float* __restrict__ eps_z0, const float* __restrict__ eps_th,
              const float* __restrict__ gz_w, const float* __restrict__ gz_b,
              const float* __restrict__ gth_w, const float* __restrict__ gth_b,
              float* __restrict__ o_mu_z0, float* __restrict__ o_lv_z0,
              float* __restrict__ o_mu_th, float* __restrict__ o_lv_th,
              float* __restrict__ z0_hat, float* __restrict__ theta_hat) {
    const int tid = threadIdx.x;
    __shared__ float z0s[BB][ZD];
    __shared__ float ths[BB][THD];

    for (int idx = tid; idx < BB * ZD; idx += 256) {
        int b = idx >> 4, o = idx & 15;
        const float* pr = pooled + (size_t)b * EH;
        float mu = b_mu_z0[o], lv = b_lv_z0[o];
        for (int k = 0; k < EH; ++k) {
            float p = pr[k];
            mu += p * w_mu_z0[k * ZD + o];
            lv += p * w_lv_z0[k * ZD + o];
        }
        o_mu_z0[idx] = mu; o_lv_z0[idx] = lv;
        z0s[b][o] = mu + eps_z0[idx] * expf(0.5f * lv);
    }
    for (int idx = tid; idx < BB * THD; idx += 256) {
        int b = idx >> 2, o = idx & 3;
        const float* pr = pooled + (size_t)b * EH;
        float mu = b_mu_th[o], lv = b_lv_th[o];
        for (int k = 0; k < EH; ++k) {
            float p = pr[k];
            mu += p * w_mu_th[k * THD + o];
            lv += p * w_lv_th[k * THD + o];
        }
        o_mu_th[idx] = mu; o_lv_th[idx] = lv;
        ths[b][o] = mu + eps_th[idx] * expf(0.5f * lv);
    }
    __syncthreads();
    for (int idx = tid; idx < BB * ZD; idx += 256) {
        int b = idx >> 4, o = idx & 15;
        float s = gz_b[o];
        for (int k = 0; k < ZD; ++k) s += z0s[b][k] * gz_w[k * ZD + o];
        z0_hat[idx] = s;
    }
    for (int idx = tid; idx < BB * THD; idx += 256) {
        int b = idx >> 2, o = idx & 3;
        float s = gth_b[o];
        for (int k = 0; k < THD; ++k) s += ths[b][k] * gth_w[k * THD + o];
        theta_hat[idx] = (s > 20.0f) ? s : log1pf(expf(s));   // softplus
    }
}

// ---------------------------------------------------------------------------
// Kernel 3: fixed-step RK4, one workgroup per batch element; trajectory,
// ode_w2, actions resident in LDS, ode_w1 columns in registers.
// ---------------------------------------------------------------------------
__global__ __launch_bounds__(128)
void k_rk4(const float* __restrict__ tp, const float* __restrict__ acts_g,
           const float* __restrict__ z0_hat, const float* __restrict__ theta_hat,
           const float* __restrict__ w1, const float* __restrict__ b1,
           const float* __restrict__ w2, const float* __restrict__ b2g,
           float* __restrict__ zhat_out) {
    const int b = blockIdx.x;
    const int tid = threadIdx.x;

    __shared__ float traj[NSTEPS + 1][ZD];
    __shared__ float harr[OHID];
    __shared__ float w2s[OHID][ZD];
    __shared__ float acts[TT][AD];
    __shared__ float thh[THD];
    __shared__ float b2s[ZD];
    __shared__ float zt[ZD];
    __shared__ float ks[4][ZD];

    float w1r[ZD + THD + AD];
    #pragma unroll
    for (int i = 0; i < ZD + THD + AD; ++i) w1r[i] = w1[i * OHID + tid];
    const float b1r = b1[tid];
    #pragma unroll
    for (int o = 0; o < ZD; ++o) w2s[tid][o] = w2[tid * ZD + o];
    if (tid < TT * AD) ((float*)acts)[tid] = acts_g[(size_t)b * TT * AD + tid];
    if (tid < THD) thh[tid] = theta_hat[b * THD + tid];
    if (tid < ZD) { b2s[tid] = b2g[tid]; traj[0][tid] = z0_hat[b * ZD + tid]; }

    const float t0 = tp[0], t1 = tp[TT - 1];
    const float hstep = (t1 - t0) / (float)NSTEPS;
    const float inv = 1.0f / (t1 - t0);
    __syncthreads();

    auto feval = [&](float tt, const float* zin, float* kout) {
        float frac = (tt - t0) * inv;
        int fi = (int)roundf(frac * (float)(TT - 1));
        fi = min(max(fi, 0), TT - 1);
        float u0 = acts[fi][0], u1 = acts[fi][1];
        float hj = b1r;
        #pragma unroll
        for (int q = 0; q < ZD; ++q) hj += zin[q] * w1r[q];
        #pragma unroll
        for (int q = 0; q < THD; ++q) hj += thh[q] * w1r[ZD + q];
        hj += u0 * w1r[ZD + THD] + u1 * w1r[ZD + THD + 1];
        harr[tid] = tanhf(hj);
        __syncthreads();
        if (tid < ZD) {
            float s = b2s[tid];
            for (int j = 0; j < OHID; ++j) s += harr[j] * w2s[j][tid];
            kout[tid] = s;
        }
        __syncthreads();
    };

    for (int i = 0; i < NSTEPS; ++i) {
        float t = t0 + (float)i * hstep;
        feval(t, traj[i], ks[0]);
        if (tid < ZD) zt[tid] = traj[i][tid] + 0.5f * hstep * ks[0][tid];
        __syncthreads();
        feval(t + 0.5f * hstep, zt, ks[1]);
        if (tid < ZD) zt[tid] = traj[i][tid] + 0.5f * hstep * ks[1][tid];
        __syncthreads();
        feval(t + 0.5f * hstep, zt, ks[2]);
        if (tid < ZD) zt[tid] = traj[i][tid] + hstep * ks[2][tid];
        __syncthreads();
        feval(t + hstep, zt, ks[3]);
        if (tid < ZD)
            traj[i + 1][tid] = traj[i][tid] + (hstep / 6.0f) *
                (ks[0][tid] + 2.0f * ks[1][tid] + 2.0f * ks[2][tid] + ks[3][tid]);
        __syncthreads();
    }

    // linear interpolation at the requested output times
    for (int idx = tid; idx < TT * ZD; idx += 128) {
        int t = idx >> 4, o = idx & 15;
        float pos = (tp[t] - t0) * inv * (float)NSTEPS;
        int i0 = (int)floorf(pos);
        i0 = min(max(i0, 0), NSTEPS - 1);
        float wgt = pos - (float)i0;
        zhat_out[((size_t)b * TT + t) * ZD + o] =
            traj[i0][o] * (1.0f - wgt) + traj[i0 + 1][o] * wgt;
    }
}

// ---------------------------------------------------------------------------
// Kernel 4: decoder hidden layer (K=16, trivial).
// ---------------------------------------------------------------------------
__global__ __launch_bounds__(256)
void k_dec_hidden(const float* __restrict__ zhat, const float* __restrict__ w1,
                  const float* __restrict__ b1, float* __restrict__ dhout) {
    const int row = blockIdx.x;
    const int j = threadIdx.x;
    __shared__ float zr[ZD];
    if (j < ZD) zr[j] = zhat[(size_t)row * ZD + j];
    __syncthreads();
    float s = b1[j];
    #pragma unroll
    for (int z = 0; z < ZD; ++z) s += zr[z] * w1[z * DH + j];
    dhout[(size_t)row * DH + j] = tanhf(s);
}

// ---------------------------------------------------------------------------
// Kernel 5: decoder output GEMM [3200x256]@[256x16384] + sigmoid, fp32 WMMA,
// double-buffered async staging. grid = (256 n-tiles, 50 m-tiles), block 256.
// ---------------------------------------------------------------------------
__global__ __launch_bounds__(256)
void k_decoder(const float* __restrict__ dhm, const float* __restrict__ w,
               const float* __restrict__ bias, float* __restrict__ out) {
    const int nbase = blockIdx.x * 64;
    const int mbase = blockIdx.y * 64;
    const int tid   = threadIdx.x;
    const int lane  = tid & 31;
    const int wv    = tid >> 5;
    const int mtile = wv & 3;
    const int npair = wv >> 2;

    __shared__ float lA[2][64][36];
    __shared__ float lB[2][32][68];

    v8f acc0 = {}; v8f acc1 = {};

    const int tA = tid >> 2, qA = tid & 3;
    const int rB = tid >> 4, cB = tid & 15;
    const float* gA = dhm + (size_t)(mbase + tA) * DH + qA * 8;

    auto stage = [&](int k0, int bf) {
        cp_async16(gA + k0,     &lA[bf][tA][qA * 8]);
        cp_async16(gA + k0 + 4, &lA[bf][tA][qA * 8 + 4]);
        cp_async16(w + (size_t)(k0 + rB) * IMG + nbase + cB * 4,      &lB[bf][rB][cB * 4]);
        cp_async16(w + (size_t)(k0 + rB + 16) * IMG + nbase + cB * 4, &lB[bf][rB + 16][cB * 4]);
    };

    const int kb   = (lane >> 4) * 2;
    const int mrow = mtile * 16 + (lane & 15);
    const int nc0  = npair * 32 + (lane & 15);

    stage(0, 0);
    const int NCH = DH / KC;
    for (int c = 0; c < NCH; ++c) {
        const int bf = c & 1;
        if (c + 1 < NCH) { stage((c + 1) * KC, bf ^ 1); WAIT_ASYNC(4); }
        else             { WAIT_ASYNC(0); }
        __syncthreads();
        #pragma unroll
        for (int kk = 0; kk < KC; kk += 4) {
            v2f a  = *(const v2f*)&lA[bf][mrow][kk + kb];
            v2f b0 = { lB[bf][kk + kb][nc0],      lB[bf][kk + kb + 1][nc0]      };
            v2f b1 = { lB[bf][kk + kb][nc0 + 16], lB[bf][kk + kb + 1][nc0 + 16] };
            acc0 = wmma4(a, b0, acc0);
            acc1 = wmma4(a, b1, acc1);
        }
        __syncthreads();
    }

    const int half = lane >> 4;
    const int r0   = mbase + mtile * 16 + half * 8;
    const int c0   = nbase + npair * 32 + (lane & 15);
    const int c1   = c0 + 16;
    const float bb0 = bias[c0], bb1 = bias[c1];
    #pragma unroll
    for (int v = 0; v < 8; ++v) {
        float s0 = acc0[v] + bb0;
        float s1 = acc1[v] + bb1;
        out[(size_t)(r0 + v) * IMG + c0] = 1.0f / (1.0f + expf(-s0));
        out[(size_t)(r0 + v) * IMG + c1] = 1.0f / (1.0f + expf(-s1));
    }
}

// ---------------------------------------------------------------------------
extern "C" void kernel_launch(void* const* d_in, const int* in_sizes, int n_in,
                              void* d_out, int out_size, void* d_ws, size_t ws_size,
                              hipStream_t stream) {
    const float* x_seq    = (const float*)d_in[0];
    const float* t_points = (const float*)d_in[1];
    const float* actions  = (const float*)d_in[2];
    const float* eps_z0   = (const float*)d_in[3];
    const float* eps_th   = (const float*)d_in[4];
    const float* enc_w1   = (const float*)d_in[5];
    const float* enc_b1   = (const float*)d_in[6];
    const float* w_mu_z0  = (const float*)d_in[7];
    const float* b_mu_z0  = (const float*)d_in[8];
    const float* w_lv_z0  = (const float*)d_in[9];
    const float* b_lv_z0  = (const float*)d_in[10];
    const float* w_mu_th  = (const float*)d_in[11];
    const float* b_mu_th  = (const float*)d_in[12];
    const float* w_lv_th  = (const float*)d_in[13];
    const float* b_lv_th  = (const float*)d_in[14];
    const float* gz_w     = (const float*)d_in[15];
    const float* gz_b     = (const float*)d_in[16];
    const float* gth_w    = (const float*)d_in[17];
    const float* gth_b    = (const float*)d_in[18];
    const float* ode_w1   = (const float*)d_in[19];
    const float* ode_b1   = (const float*)d_in[20];
    const float* ode_w2   = (const float*)d_in[21];
    const float* ode_b2   = (const float*)d_in[22];
    const float* dec_w1   = (const float*)d_in[23];
    const float* dec_b1   = (const float*)d_in[24];
    const float* dec_w2   = (const float*)d_in[25];
    const float* dec_b2   = (const float*)d_in[26];

    float* out = (float*)d_out;
    float* out_xhat  = out;                                     // [64,50,16384]
    float* out_mu_z0 = out + (size_t)BB * TT * IMG;             // [64,16]
    float* out_lv_z0 = out_mu_z0 + BB * ZD;                     // [64,16]
    float* out_mu_th = out_lv_z0 + BB * ZD;                     // [64,4]
    float* out_lv_th = out_mu_th + BB * THD;                    // [64,4]
    float* out_zhat  = out_lv_th + BB * THD;                    // [64,50,16]

    float* ws        = (float*)d_ws;
    float* pooled    = ws;                                      // 64*256
    float* z0_hat    = pooled + BB * EH;                        // 64*16
    float* theta_hat = z0_hat + BB * ZD;                        // 64*4
    float* dh        = theta_hat + BB * THD;                    // 3200*256

    k_encoder<<<dim3(4, BB), 256, 0, stream>>>(x_seq, enc_w1, enc_b1, pooled);
    k_latent<<<1, 256, 0, stream>>>(pooled, w_mu_z0, b_mu_z0, w_lv_z0, b_lv_z0,
                                    w_mu_th, b_mu_th, w_lv_th, b_lv_th,
                                    eps_z0, eps_th, gz_w, gz_b, gth_w, gth_b,
                                    out_mu_z0, out_lv_z0, out_mu_th, out_lv_th,
                                    z0_hat, theta_hat);
    k_rk4<<<BB, 128, 0, stream>>>(t_points, actions, z0_hat, theta_hat,
                                  ode_w1, ode_b1, ode_w2, ode_b2, out_zhat);
    k_dec_hidden<<<BB * TT, 256, 0, stream>>>(out_zhat, dec_w1, dec_b1, dh);
    k_decoder<<<dim3(IMG / 64, BB * TT / 64), 256, 0, stream>>>(dh, dec_w2, dec_b2, out_xhat);
}